// YOLOv2Loss_63445256896605
// MI455X (gfx1250) — compile-verified
//
#include <hip/hip_runtime.h>
#include <math.h>

// Problem constants (match reference)
#define NA 5
#define NC 80
#define FH 19
#define FW 19
#define NT 50
#define HW (FH*FW)              // 361
#define CELLS_PER_B (NA*HW)     // 1805
#define NOOBJ_SCALE 1.0f
#define OBJ_SCALE   5.0f
#define BG_THRESH   0.6f

typedef __attribute__((ext_vector_type(2))) float v2f;
typedef __attribute__((ext_vector_type(8))) float v8f;

// ---------------------------------------------------------------------------
// Wave32 sum using V_WMMA_F32_16X16X4_F32 (exact f32 FMA path).
// Lane L (0..15) contributes A[L,0]; lane L (16..31) contributes A[L-16,2];
// A[:,1]=A[:,3]=0. B = all-ones 4x16  =>  D[m,n] = acc_m + acc_{m+16}.
// Per-lane sum of the 8 D VGPRs = sum of rows 0..7 (lanes 0-15) or rows 8..15
// (lanes 16-31); total = value@lane0 + value@lane16.
// EXEC must be all-1s at the call site: call outside any divergent region.
// ---------------------------------------------------------------------------
__device__ __forceinline__ float wave_sum32(float acc) {
    v2f a; a[0] = acc;  a[1] = 0.0f;
    v2f b; b[0] = 1.0f; b[1] = 1.0f;
    v8f c = {};
    v8f d = __builtin_amdgcn_wmma_f32_16x16x4_f32(
        /*neg_a=*/false, a, /*neg_b=*/false, b,
        /*c_mod=*/(short)0, c, /*reuse_a=*/false, /*reuse_b=*/false);
    float s = d[0] + d[1] + d[2] + d[3] + d[4] + d[5] + d[6] + d[7];
    return __shfl(s, 0, 32) + __shfl(s, 16, 32);
}

__device__ __forceinline__ float iou_cwh(float ax, float ay, float aw, float ah,
                                         float bx, float by, float bw, float bh) {
    float ax1 = ax - aw * 0.5f, ax2 = ax + aw * 0.5f;
    float ay1 = ay - ah * 0.5f, ay2 = ay + ah * 0.5f;
    float bx1 = bx - bw * 0.5f, bx2 = bx + bw * 0.5f;
    float by1 = by - bh * 0.5f, by2 = by + bh * 0.5f;
    float uw = fmaxf(ax2, bx2) - fminf(ax1, bx1);
    float uh = fmaxf(ay2, by2) - fminf(ay1, by1);
    float cw = aw + bw - uw;
    float ch = ah + bh - uh;
    float inter = (cw > 0.0f && ch > 0.0f) ? cw * ch : 0.0f;
    float uni = aw * ah + bw * bh - inter;
    return inter / fmaxf(uni, 1e-12f);
}

__device__ __forceinline__ float sigmoidf_(float v) { return 1.0f / (1.0f + expf(-v)); }

// ---------------------------------------------------------------------------
// Kernel 0: reset winner map (workspace is poisoned between runs)
// ---------------------------------------------------------------------------
__global__ void k_init_winner(int* __restrict__ winner, int n) {
    int i = blockIdx.x * blockDim.x + threadIdx.x;
    if (i < n) winner[i] = -1;
}

// ---------------------------------------------------------------------------
// Kernel 1: per-target prep. grid = B blocks, 64 threads.
// rec[b][t][12] = {tx,ty,tw,th,cls, gx,gy,gw,gh, pad,pad,pad}
// gtbox[b][t][4] = {gx,gy,gw,gh}
// winner[cell] = max valid t scattering into that cell (matches last-wins)
// ---------------------------------------------------------------------------
__global__ void k_prep_targets(const float* __restrict__ target,
                               const float* __restrict__ anchors,
                               int*   __restrict__ winner,
                               int*   __restrict__ nvalid,
                               float* __restrict__ gtbox,
                               float* __restrict__ rec) {
    int b = blockIdx.x;
    const float* tb = target + (size_t)b * NT * 5;
    __shared__ int s_nv;
    if (threadIdx.x == 0) {
        int nv = 0;
        while (nv < NT && tb[nv * 5 + 1] != 0.0f) ++nv;  // cumprod(x != 0)
        s_nv = nv;
        nvalid[b] = nv;
    }
    __syncthreads();
    int t = threadIdx.x;
    if (t < s_nv) {
        float cls = tb[t * 5 + 0];
        float gx = tb[t * 5 + 1] * (float)FW;
        float gy = tb[t * 5 + 2] * (float)FH;
        float gw = tb[t * 5 + 3] * (float)FW;
        float gh = tb[t * 5 + 4] * (float)FH;
        // best anchor by anchor-space IoU (first max wins, like jnp.argmax)
        int bn = 0; float best = -1.0f;
        #pragma unroll
        for (int a = 0; a < NA; ++a) {
            float aw = anchors[a * 2 + 0], ah = anchors[a * 2 + 1];
            float inter = fminf(gw, aw) * fminf(gh, ah);
            float uni   = gw * gh + aw * ah - inter;
            float r = inter / fmaxf(uni, 1e-12f);
            if (r > best) { best = r; bn = a; }
        }
        int gi = (int)gx; gi = gi < 0 ? 0 : (gi > FW - 1 ? FW - 1 : gi);
        int gj = (int)gy; gj = gj < 0 ? 0 : (gj > FH - 1 ? FH - 1 : gj);
        float* rp = rec + ((size_t)b * NT + t) * 12;
        rp[0] = gx - (float)gi;
        rp[1] = gy - (float)gj;
        rp[2] = logf(fmaxf(gw, 1e-12f) / anchors[bn * 2 + 0]);
        rp[3] = logf(fmaxf(gh, 1e-12f) / anchors[bn * 2 + 1]);
        rp[4] = cls;
        rp[5] = gx; rp[6] = gy; rp[7] = gw; rp[8] = gh;
        float* gp = gtbox + ((size_t)b * NT + t) * 4;
        gp[0] = gx; gp[1] = gy; gp[2] = gw; gp[3] = gh;
        int cell = b * CELLS_PER_B + bn * HW + gj * FW + gi;
        atomicMax(&winner[cell], t);
    }
}

// ---------------------------------------------------------------------------
// Kernel 2: per-cell fused loss. grid = (8, B), block = 256.
// Only 5 of the 85 channel planes are streamed; class logits are gathered
// solely at winner cells (~B*T of 115k cells).
// ---------------------------------------------------------------------------
__global__ void k_cell_loss(const float* __restrict__ outp,
                            const float* __restrict__ anchors,
                            const int*   __restrict__ winner,
                            const int*   __restrict__ nvalid,
                            const float* __restrict__ gtbox,
                            const float* __restrict__ rec,
                            float* __restrict__ partials) {
    int b   = blockIdx.y;
    int cid = blockIdx.x * blockDim.x + threadIdx.x;

    __shared__ float sgt[NT * 4];
    __shared__ int   s_nv;
    __shared__ float wpart[8];

    if (threadIdx.x == 0) s_nv = nvalid[b];
    if (threadIdx.x < NT * 4) sgt[threadIdx.x] = gtbox[(size_t)b * NT * 4 + threadIdx.x];
    __syncthreads();

    float acc = 0.0f;
    if (cid < CELLS_PER_B) {
        int a = cid / HW;
        int r = cid - a * HW;          // j*FW + i
        int j = r / FW;
        int i = r - j * FW;

        const float* cp = outp + ((size_t)(b * NA + a) * (5 + NC)) * HW + r;
        float ox = cp[0 * HW];
        float oy = cp[1 * HW];
        float ow = cp[2 * HW];
        float oh = cp[3 * HW];
        float oc = cp[4 * HW];

        float x    = sigmoidf_(ox);
        float y    = sigmoidf_(oy);
        float conf = sigmoidf_(oc);
        float aw = anchors[a * 2 + 0], ah = anchors[a * 2 + 1];
        float px = x + (float)i, py = y + (float)j;
        float pw = expf(ow) * aw,  ph = expf(oh) * ah;

        // background: best IoU vs all valid gt boxes (LDS-cached)
        float best = 0.0f;
        int nv = s_nv;
        for (int t = 0; t < nv; ++t) {
            best = fmaxf(best, iou_cwh(px, py, pw, ph,
                                       sgt[t * 4 + 0], sgt[t * 4 + 1],
                                       sgt[t * 4 + 2], sgt[t * 4 + 3]));
        }
        float conf_mask = (best > BG_THRESH) ? 0.0f : NOOBJ_SCALE;

        float tconf = 0.0f;
        int wt = winner[(size_t)b * CELLS_PER_B + cid];
        if (wt >= 0) {
            const float* lp = cp + 5 * HW;   // class logits for this cell
            __builtin_prefetch(lp, 0, 0);    // global_prefetch_b8
            const float* rp = rec + ((size_t)b * NT + wt) * 12;
            float tx = rp[0], ty = rp[1], tw = rp[2], th = rp[3];
            int   ci = (int)rp[4];
            tconf = iou_cwh(rp[5], rp[6], rp[7], rp[8], px, py, pw, ph);
            conf_mask = OBJ_SCALE;
            float dx = x - tx, dy = y - ty, dw = ow - tw, dh = oh - th;
            acc += 0.5f * (dx * dx + dy * dy + dw * dw + dh * dh);
            // cross-entropy over 80 classes: two strided passes (L1-hot 2nd pass)
            float m = -1e30f;
            for (int c = 0; c < NC; ++c) m = fmaxf(m, lp[c * HW]);
            float se = 0.0f, lc = 0.0f;
            for (int c = 0; c < NC; ++c) {
                float v = lp[c * HW];
                se += expf(v - m);
                if (c == ci) lc = v;
            }
            acc += -(lc - m - logf(se));     // CLASS_SCALE = 1
        }
        float dc = conf - tconf;
        acc += 0.5f * conf_mask * dc * dc;   // (sqrt(mask)*(..))^2 == mask*(..)^2
    }

    // block reduction (uniform control flow: EXEC all-1s for WMMA)
    float wsum = wave_sum32(acc);
    int lane = threadIdx.x & 31, wid = threadIdx.x >> 5;
    if (lane == 0) wpart[wid] = wsum;
    __syncthreads();
    if (threadIdx.x == 0) {
        float s = 0.0f;
        #pragma unroll
        for (int k = 0; k < 8; ++k) s += wpart[k];
        partials[(size_t)b * gridDim.x + blockIdx.x] = s;
    }
}

// ---------------------------------------------------------------------------
// Kernel 3: deterministic final reduction of block partials -> scalar loss
// ---------------------------------------------------------------------------
__global__ void k_final_reduce(const float* __restrict__ partials, int n,
                               float* __restrict__ out) {
    __shared__ float wpart[8];
    int tid = threadIdx.x;
    float v = 0.0f;
    for (int i = tid; i < n; i += 256) v += partials[i];
    float wsum = wave_sum32(v);
    int lane = tid & 31, wid = tid >> 5;
    if (lane == 0) wpart[wid] = wsum;
    __syncthreads();
    if (tid == 0) {
        float s = 0.0f;
        #pragma unroll
        for (int k = 0; k < 8; ++k) s += wpart[k];
        out[0] = s;
    }
}

// ---------------------------------------------------------------------------
extern "C" void kernel_launch(void* const* d_in, const int* in_sizes, int n_in,
                              void* d_out, int out_size, void* d_ws, size_t ws_size,
                              hipStream_t stream) {
    const float* outp    = (const float*)d_in[0];  // (B, 425, 19, 19)
    const float* target  = (const float*)d_in[1];  // (B, 250)
    const float* anchors = (const float*)d_in[2];  // (5, 2)

    const int B = in_sizes[1] / (NT * 5);
    const int ncells = B * CELLS_PER_B;
    const int gx = (CELLS_PER_B + 255) / 256;      // 8 blocks of cells per batch
    const int npart = B * gx;

    // Workspace layout (bytes)
    char* ws = (char*)d_ws;
    size_t off = 0;
    int*   winner = (int*)(ws + off);  off += (size_t)ncells * sizeof(int);
    int*   nvalid = (int*)(ws + off);  off += ((size_t)B * sizeof(int) + 255) & ~(size_t)255;
    float* gtbox  = (float*)(ws + off); off += (size_t)B * NT * 4 * sizeof(float);
    float* rec    = (float*)(ws + off); off += (size_t)B * NT * 12 * sizeof(float);
    float* partials = (float*)(ws + off); off += (size_t)npart * sizeof(float);
    (void)ws_size; (void)n_in; (void)out_size;

    k_init_winner<<<(ncells + 255) / 256, 256, 0, stream>>>(winner, ncells);
    k_prep_targets<<<B, 64, 0, stream>>>(target, anchors, winner, nvalid, gtbox, rec);
    k_cell_loss<<<dim3(gx, B), 256, 0, stream>>>(outp, anchors, winner, nvalid,
                                                 gtbox, rec, partials);
    k_final_reduce<<<1, 256, 0, stream>>>(partials, npart, (float*)d_out);
}